// VQModel_70669391888510
// MI455X (gfx1250) — compile-verified
//
#include <hip/hip_runtime.h>
#include <cstdint>

// Problem sizes (fixed by the reference)
#define NROWS  32768
#define KCODES 8192
#define DDIM   64
#define WIDTH  128

#define HSTR   132          // padded stride for 16x128 activation tiles (conflict-free frag reads)
#define ESTR   68           // padded stride for staged codebook rows
#define CHUNK  128          // codes staged per LDS chunk (32KB payload)
#define NCHUNK (KCODES / CHUNK)
#define EBYTES (CHUNK * ESTR * 4)

typedef __attribute__((ext_vector_type(2))) float v2f;
typedef __attribute__((ext_vector_type(8))) float v8f;

// Explicit global address-space pointers: force global_load/global_store (never FLAT).
typedef const __attribute__((address_space(1))) float* gfp;
typedef __attribute__((address_space(1))) float*       gfpw;
typedef const __attribute__((address_space(1))) v2f*   gv2p;
typedef __attribute__((address_space(1))) v2f*         gv2pw;

static __device__ __forceinline__ v8f splat8(float v) {
  v8f r;
#pragma unroll
  for (int i = 0; i < 8; ++i) r[i] = v;
  return r;
}

static __device__ __forceinline__ v8f wmma_f32(v2f a, v2f b, v8f c) {
  // V_WMMA_F32_16X16X4_F32 : D = A(16x4) x B(4x16) + C(16x16), all f32
  return __builtin_amdgcn_wmma_f32_16x16x4_f32(false, a, false, b, (short)0, c, false, false);
}

static __device__ __forceinline__ uint32_t lds_addr32(const void* p) {
  return (uint32_t)(uintptr_t)p;   // low 32 bits of flat address == LDS offset
}

// CDNA5 async global->LDS copy (ASYNCcnt-tracked), GV addressing mode.
static __device__ __forceinline__ void async_ld_b128(uint32_t lds, const void* g) {
  asm volatile("global_load_async_to_lds_b128 %0, %1, off" :: "v"(lds), "v"(g) : "memory");
}
static __device__ __forceinline__ void wait_async0() {
  asm volatile("s_wait_asynccnt 0x0" ::: "memory");
}
static __device__ __forceinline__ void wait_async8() {
  asm volatile("s_wait_asynccnt 0x8" ::: "memory");
}

// Stage one 128-code chunk (2048 x 16B). Per thread: 8 async b128 loads; the
// 32 lanes of a wave cover a contiguous 512B span per load -> fully coalesced.
// gchunk = this thread's first 16B unit inside the chunk; ldsBase matches it.
static __device__ __forceinline__ void stage_chunk(const float* gchunk, uint32_t ldsBase) {
#pragma unroll
  for (int i = 0; i < 8; ++i)
    async_ld_b128(ldsBase + i * (16 * ESTR * 4), (const void*)(gchunk + i * 16 * DDIM));
}

// One 16x16 output tile of a [16 x 128] @ [128 x ntile*16] dense layer.
// B fragments preloaded into registers (global, clause-able); dual accumulator
// chains for WMMA ILP; A fragments from per-wave LDS tile.
static __device__ __forceinline__ v8f dense_tile(gfp Wg, float bias, const float* hwave,
                                                 int laneM, int kOff, int t, bool do_relu) {
  v2f bf[32];
#pragma unroll
  for (int s = 0; s < 32; ++s)
    bf[s] = *(gv2p)&Wg[(size_t)(t * 16 + laneM) * WIDTH + 4 * s + kOff];
  v8f a0 = splat8(bias), a1 = splat8(0.0f);
#pragma unroll
  for (int s = 0; s < 32; s += 2) {
    v2f af0 = *(const v2f*)&hwave[laneM * HSTR + 4 * s + kOff];
    v2f af1 = *(const v2f*)&hwave[laneM * HSTR + 4 * (s + 1) + kOff];
    a0 = wmma_f32(af0, bf[s], a0);
    a1 = wmma_f32(af1, bf[s + 1], a1);
  }
  v8f r = a0 + a1;
  if (do_relu) {
#pragma unroll
    for (int j = 0; j < 8; ++j) r[j] = fmaxf(r[j], 0.0f);
  }
  return r;
}

// Full 128 -> 128 layer with relu, LDS tile updated in place (reads complete
// before writes since all 8 tiles are buffered in registers first).
static __device__ __forceinline__ void layer128(gfp Wg, gfp bg, float* hwave,
                                                int laneM, int laneH, int kOff) {
  v8f acc[8];
  for (int t = 0; t < 8; ++t)
    acc[t] = dense_tile(Wg, bg[t * 16 + laneM], hwave, laneM, kOff, t, true);
  for (int t = 0; t < 8; ++t)
#pragma unroll
    for (int j = 0; j < 8; ++j)
      hwave[(j + 8 * laneH) * HSTR + t * 16 + laneM] = acc[t][j];
}

// ---------------- kernel 1: -0.5 * ||E_c||^2 per code ----------------
__global__ __launch_bounds__(256) void vq_code_norms(const float* __restrict__ E,
                                                     float* __restrict__ nrm) {
  int wave = threadIdx.x >> 5, lane = threadIdx.x & 31;
  int c = blockIdx.x * 8 + wave;
  v2f e2 = *(gv2p)&((gfp)E)[(size_t)c * DDIM + 2 * lane];
  float s = e2.x * e2.x + e2.y * e2.y;
#pragma unroll
  for (int m = 1; m < 32; m <<= 1) s += __shfl_xor(s, m, 32);
  if (lane == 0) ((gfpw)nrm)[c] = -0.5f * s;
}

// ---------------- kernel 2: fused MLP + VQ argmin + gather + decode ----------------
// out layout: [N] out | [N] hard_idx (as float) | [N*64] z_e | [N*64] z_q
__global__ __launch_bounds__(256) void vq_fused(
    const float* __restrict__ x,
    const float* __restrict__ W1, const float* __restrict__ b1,
    const float* __restrict__ W2, const float* __restrict__ b2,
    const float* __restrict__ W3, const float* __restrict__ b3,
    const float* __restrict__ W4, const float* __restrict__ b4,
    const float* __restrict__ E,
    const float* __restrict__ decW, const float* __restrict__ decb,
    const float* __restrict__ negHalfNorm,
    float* __restrict__ out) {
  __shared__ float hbuf[8][16 * HSTR];       // per-wave activation tile (16 rows x 128, padded)
  __shared__ float ebuf[2][CHUNK * ESTR];    // double-buffered codebook chunk
  __shared__ int   bidx[8][16];

  const int tid   = threadIdx.x;
  const int wave  = tid >> 5;
  const int lane  = tid & 31;
  const int laneM = lane & 15;
  const int laneH = lane >> 4;
  const int kOff  = laneH * 2;               // f32 A/B fragment K-offset (16x16x4 layout)
  const int row0  = blockIdx.x * 128 + wave * 16;
  float* hw = &hbuf[wave][0];

  // Per-thread staging state: slot within a chunk + matching LDS base (buffer 0).
  const float*   gsE   = E + (size_t)(tid >> 4) * DDIM + (tid & 15) * 4;
  const uint32_t ldsE0 = lds_addr32(&ebuf[0][(tid >> 4) * ESTR + (tid & 15) * 4]);

  // Kick chunk 0 while the MLP runs.
  stage_chunk(gsE, ldsE0);

  // ---- layer 1: 1 -> 128 (elementwise) ----
  float xv = ((gfp)x)[row0 + laneM];
  {
    gfp W1g = (gfp)W1, b1g = (gfp)b1;
    for (int j = laneH; j < WIDTH; j += 2)
      hw[laneM * HSTR + j] = fmaxf(xv * W1g[j] + b1g[j], 0.0f);
  }

  // ---- layers 2,3: 128 -> 128 + relu (WMMA f32 16x16x4) ----
  layer128((gfp)W2, (gfp)b2, hw, laneM, laneH, kOff);
  layer128((gfp)W3, (gfp)b3, hw, laneM, laneH, kOff);

  // ---- layer 4: 128 -> 64 (z_e) ----
  v8f zacc[4];
  {
    gfp W4g = (gfp)W4, b4g = (gfp)b4;
    for (int t = 0; t < 4; ++t)
      zacc[t] = dense_tile(W4g, b4g[t * 16 + laneM], hw, laneM, kOff, t, false);
  }
  gfpw zeg = (gfpw)(out + (size_t)2 * NROWS);
  for (int t = 0; t < 4; ++t)
#pragma unroll
    for (int j = 0; j < 8; ++j) {
      int m = j + 8 * laneH;
      hw[m * HSTR + t * 16 + laneM] = zacc[t][j];
      zeg[(size_t)(row0 + m) * DDIM + t * 16 + laneM] = zacc[t][j];
    }

  // A-fragments of z_e, kept in registers for all 512 code tiles.
  v2f za[16];
#pragma unroll
  for (int s = 0; s < 16; ++s)
    za[s] = *(const v2f*)&hw[laneM * HSTR + 4 * s + kOff];

  // ---- fused distance GEMM + running argmax of (z.E_c - 0.5||E_c||^2) ----
  gfp nrmg = (gfp)negHalfNorm;
  float bestV[8];
  int   bestI[8];
#pragma unroll
  for (int j = 0; j < 8; ++j) { bestV[j] = -3.4e38f; bestI[j] = 0; }

  for (int c = 0; c < NCHUNK; ++c) {
    if (c + 1 < NCHUNK) {
      stage_chunk(gsE + (size_t)(c + 1) * CHUNK * DDIM, ldsE0 + ((c + 1) & 1) * EBYTES);
      wait_async8();             // chunk c complete (async loads retire in order)
    } else {
      wait_async0();
    }
    __syncthreads();
    const float* ebp = &ebuf[c & 1][0];
    for (int t = 0; t < 8; ++t) {
      int col = c * CHUNK + t * 16 + laneM;
      v2f bf[16];
#pragma unroll
      for (int s = 0; s < 16; ++s)
        bf[s] = *(const v2f*)&ebp[(t * 16 + laneM) * ESTR + 4 * s + kOff];
      v8f a0 = splat8(nrmg[col]), a1 = splat8(0.0f);
#pragma unroll
      for (int s = 0; s < 16; s += 2) {
        a0 = wmma_f32(za[s],     bf[s],     a0);
        a1 = wmma_f32(za[s + 1], bf[s + 1], a1);
      }
      v8f a8 = a0 + a1;
#pragma unroll
      for (int j = 0; j < 8; ++j)
        if (a8[j] > bestV[j]) { bestV[j] = a8[j]; bestI[j] = col; }
    }
    __syncthreads();             // WAR: everyone done reading before this buffer is restaged
  }

  // reduce argmax across the 16 lanes of each half (rows 0-7 / 8-15), first-index tie-break
#pragma unroll
  for (int m = 1; m < 16; m <<= 1) {
#pragma unroll
    for (int j = 0; j < 8; ++j) {
      float ov = __shfl_xor(bestV[j], m, 32);
      int   oi = __shfl_xor(bestI[j], m, 32);
      if (ov > bestV[j] || (ov == bestV[j] && oi < bestI[j])) { bestV[j] = ov; bestI[j] = oi; }
    }
  }
  gfpw outg = (gfpw)out;
  if (laneM == 0) {
#pragma unroll
    for (int j = 0; j < 8; ++j) {
      int m = j + 8 * laneH;
      bidx[wave][m] = bestI[j];
      outg[NROWS + row0 + m] = (float)bestI[j];   // hard_idx
    }
  }

  // ---- gather z_q and decode out = z_q . decW + decb ----
  gfpw zqg = (gfpw)(out + (size_t)2 * NROWS + (size_t)NROWS * DDIM);
  gfp  Eg  = (gfp)E;
  v2f dw = *(gv2p)&((gfp)decW)[2 * lane];
  float db = ((gfp)decb)[0];
  for (int r = 0; r < 16; ++r) {
    int idx = bidx[wave][r];
    v2f e2 = *(gv2p)&Eg[(size_t)idx * DDIM + 2 * lane];
    *(gv2pw)&zqg[(size_t)(row0 + r) * DDIM + 2 * lane] = e2;
    float p = e2.x * dw.x + e2.y * dw.y;
#pragma unroll
    for (int m2 = 1; m2 < 32; m2 <<= 1) p += __shfl_xor(p, m2, 32);
    if (lane == 0) outg[row0 + r] = p + db;
  }
}

extern "C" void kernel_launch(void* const* d_in, const int* in_sizes, int n_in,
                              void* d_out, int out_size, void* d_ws, size_t ws_size,
                              hipStream_t stream) {
  const float* x    = (const float*)d_in[0];
  const float* W1   = (const float*)d_in[1];
  const float* b1   = (const float*)d_in[2];
  const float* W2   = (const float*)d_in[3];
  const float* b2   = (const float*)d_in[4];
  const float* W3   = (const float*)d_in[5];
  const float* b3   = (const float*)d_in[6];
  const float* W4   = (const float*)d_in[7];
  const float* b4   = (const float*)d_in[8];
  const float* E    = (const float*)d_in[9];
  const float* decW = (const float*)d_in[10];
  const float* decb = (const float*)d_in[11];
  float* norms = (float*)d_ws;                 // K floats of scratch

  vq_code_norms<<<KCODES / 8, 256, 0, stream>>>(E, norms);
  vq_fused<<<NROWS / 128, 256, 0, stream>>>(x, W1, b1, W2, b2, W3, b3, W4, b4,
                                            E, decW, decb, norms, (float*)d_out);
}